// MambaBlock_74861279969328
// MI455X (gfx1250) — compile-verified
//
#include <hip/hip_runtime.h>
#include <cstddef>

// ---------------------------------------------------------------------------
// Mamba block (fp32) for gfx1250: WMMA f32 16x16x4 GEMMs with fragment-packed
// operands (b128 loads), fused conv/scan.
// ---------------------------------------------------------------------------

typedef __attribute__((ext_vector_type(2))) float v2f;
typedef __attribute__((ext_vector_type(4))) float v4f;
typedef __attribute__((ext_vector_type(8))) float v8f;

#define D_MODEL   1024
#define D_STATE   16
#define D_CONV    4
#define D_INNER   2048
#define DT_RANK   128
#define SEQ       2048
#define XDBC_COLS (DT_RANK + 2 * D_STATE)   // 160

__device__ __forceinline__ float softplus_f(float v) {
  return (v > 20.0f) ? v : log1pf(expf(v));
}
__device__ __forceinline__ float silu_f(float v) {
  return v / (1.0f + expf(-v));
}

__device__ __forceinline__ v8f wmma4(v2f a, v2f b, v8f c) {
  return __builtin_amdgcn_wmma_f32_16x16x4_f32(false, a, false, b, (short)0,
                                               c, false, false);
}

// ---------------------------------------------------------------------------
// Fragment packing.
// WMMA f32 16x16x4 A-fragment layout (ISA 7.12.2):
//   lanes 0-15  (h=0): {K=k, K=k+1} of row (mt*16 + lane)
//   lanes 16-31 (h=1): {K=k+2, K=k+3}
// We pack TWO k-steps (K=8) per lane into 4 consecutive floats:
//   Ap[((mt*K8)+k8)*128 + lane*4 + c], c -> k = k8*8 + (c>>1)*4 + h*2 + (c&1)
// B-fragment is the same permutation with col = ct*16 + (lane&15).
// ---------------------------------------------------------------------------
__global__ void pack_a_kernel(const float* __restrict__ A, int lda,
                              float* __restrict__ Ap, int K)
{
  const int idx  = blockIdx.x * blockDim.x + threadIdx.x;
  const int K8   = K >> 3;
  const int c    = idx & 3;
  const int lane = (idx >> 2) & 31;
  const int k8   = (idx >> 7) % K8;
  const int mt   = (idx >> 7) / K8;
  const int h    = lane >> 4;
  const int l16  = lane & 15;
  const int row  = mt * 16 + l16;
  const int k    = k8 * 8 + (c >> 1) * 4 + h * 2 + (c & 1);
  Ap[idx] = A[(size_t)row * lda + k];
}

__global__ void pack_b_kernel(const float* __restrict__ B, int N,
                              float* __restrict__ Bp, int K)
{
  const int idx  = blockIdx.x * blockDim.x + threadIdx.x;
  const int K8   = K >> 3;
  const int c    = idx & 3;
  const int lane = (idx >> 2) & 31;
  const int k8   = (idx >> 7) % K8;
  const int ct   = (idx >> 7) / K8;
  const int h    = lane >> 4;
  const int l16  = lane & 15;
  const int col  = ct * 16 + l16;
  const int k    = k8 * 8 + (c >> 1) * 4 + h * 2 + (c & 1);
  Bp[idx] = B[(size_t)k * N + col];
}

// ---------------------------------------------------------------------------
// Packed-operand WMMA GEMM: C[:,N] = act( A*B + bias ).
// One wave per block; each wave computes a 32 x (NT*16) strip
// (2 M-tiles x NT N-tiles, 2*NT accumulators).
// Hot loop per K=8 group: (2+NT)x global_load_b128 -> 4*NT WMMAs. Branch-free.
// ACT: 0 = none, 1 = softplus(x + bias[col])
// ---------------------------------------------------------------------------
template <int NT, int ACT>
__global__ void gemm_wmma_packed(const float* __restrict__ Ap,
                                 const float* __restrict__ Bp,
                                 float* __restrict__ C,
                                 int N, int K, int ctOff,
                                 const float* __restrict__ bias)
{
  const int lane = threadIdx.x;       // 0..31
  const int half = lane >> 4;
  const int l16  = lane & 15;
  const int mt0  = blockIdx.y * 2;            // two 16-row tiles
  const int ct0  = ctOff + blockIdx.x * NT;   // NT 16-col tiles
  const int K8   = K >> 3;

  const float* ApBase0 = Ap + (size_t)mt0 * K8 * 128 + lane * 4;
  const float* ApBase1 = ApBase0 + (size_t)K8 * 128;
  const float* BpBase  = Bp + (size_t)ct0 * K8 * 128 + lane * 4;

  v8f acc0[NT] = {};                  // rows mt0*16 .. +15
  v8f acc1[NT] = {};                  // rows mt0*16+16 .. +31

  for (int k8 = 0; k8 < K8; ++k8) {
    v4f a0 = *(const v4f*)(ApBase0 + (size_t)k8 * 128);
    v4f a1 = *(const v4f*)(ApBase1 + (size_t)k8 * 128);
    v4f b[NT];
#pragma unroll
    for (int j = 0; j < NT; ++j)
      b[j] = *(const v4f*)(BpBase + ((size_t)j * K8 + k8) * 128);

    const v2f a0lo = a0.xy, a0hi = a0.zw;
    const v2f a1lo = a1.xy, a1hi = a1.zw;
#pragma unroll
    for (int j = 0; j < NT; ++j) {
      const v2f blo = b[j].xy, bhi = b[j].zw;
      acc0[j] = wmma4(a0lo, blo, acc0[j]);
      acc1[j] = wmma4(a1lo, blo, acc1[j]);
      acc0[j] = wmma4(a0hi, bhi, acc0[j]);
      acc1[j] = wmma4(a1hi, bhi, acc1[j]);
    }
  }

  // C/D layout: VGPR r, lanes 0-15 -> M=rbase+r, lanes 16-31 -> M=rbase+r+8
  auto store = [&](v8f* acc, int rbase) {
#pragma unroll
    for (int j = 0; j < NT; ++j) {
      const int col = (ct0 + j) * 16 + l16;
      const float bv = (ACT == 1) ? bias[col] : 0.0f;
#pragma unroll
      for (int r = 0; r < 8; ++r) {
        const int row = rbase + r + half * 8;
        float v = acc[j][r];
        if (ACT == 1) v = softplus_f(v + bv);
        C[(size_t)row * N + col] = v;
      }
    }
  };
  store(acc0, mt0 * 16);
  store(acc1, mt0 * 16 + 16);
}

// ---------------------------------------------------------------------------
// Causal depthwise conv (K=4) + SiLU on the `u` half of proj.
// ---------------------------------------------------------------------------
__global__ void conv_silu_kernel(const float* __restrict__ proj,
                                 const float* __restrict__ conv_w,
                                 const float* __restrict__ conv_b,
                                 float* __restrict__ uc)
{
  const int idx = blockIdx.x * blockDim.x + threadIdx.x; // t*D_INNER + d
  const int t = idx / D_INNER;
  const int d = idx - t * D_INNER;
  float s = conv_b[d];
#pragma unroll
  for (int k = 0; k < D_CONV; ++k) {
    const int tt = t - (D_CONV - 1) + k;
    if (tt >= 0)
      s += proj[(size_t)tt * (2 * D_INNER) + d] * conv_w[d * D_CONV + k];
  }
  uc[idx] = silu_f(s);
}

// ---------------------------------------------------------------------------
// Selective scan, fused with +u*D and silu(z) gating.
// One lane per (d, n); reduction over n via __shfl_xor in 16-lane groups.
// ---------------------------------------------------------------------------
__global__ void scan_kernel(const float* __restrict__ uc,
                            const float* __restrict__ dtb,
                            const float* __restrict__ xdbc,
                            const float* __restrict__ A_log,
                            const float* __restrict__ D_param,
                            const float* __restrict__ proj,   // for z
                            float* __restrict__ yg)
{
  const int n = threadIdx.x & 15;
  const int d = blockIdx.x * (blockDim.x >> 4) + (threadIdx.x >> 4);

  const float Ad = -expf(A_log[d * D_STATE + n]);
  const float Dd = D_param[d];
  float h = 0.0f;

  for (int t = 0; t < SEQ; ++t) {
    const float dtv = dtb[(size_t)t * D_INNER + d];
    const float u   = uc [(size_t)t * D_INNER + d];
    const float Bv  = xdbc[(size_t)t * XDBC_COLS + DT_RANK + n];
    const float Cv  = xdbc[(size_t)t * XDBC_COLS + DT_RANK + D_STATE + n];

    h = expf(dtv * Ad) * h + dtv * u * Bv;

    float p = h * Cv;
    p += __shfl_xor(p, 8, 16);
    p += __shfl_xor(p, 4, 16);
    p += __shfl_xor(p, 2, 16);
    p += __shfl_xor(p, 1, 16);

    if (n == 0) {
      const float z = proj[(size_t)t * (2 * D_INNER) + D_INNER + d];
      yg[(size_t)t * D_INNER + d] = (p + u * Dd) * silu_f(z);
    }
  }
}

// ---------------------------------------------------------------------------
// Host-side orchestration.
// ---------------------------------------------------------------------------
extern "C" void kernel_launch(void* const* d_in, const int* in_sizes, int n_in,
                              void* d_out, int out_size, void* d_ws, size_t ws_size,
                              hipStream_t stream)
{
  (void)in_sizes; (void)n_in; (void)out_size; (void)ws_size;

  const float* x      = (const float*)d_in[0];
  const float* W_in   = (const float*)d_in[1];
  const float* conv_w = (const float*)d_in[2];
  const float* conv_b = (const float*)d_in[3];
  const float* W_x    = (const float*)d_in[4];
  const float* W_dt   = (const float*)d_in[5];
  const float* b_dt   = (const float*)d_in[6];
  const float* A_log  = (const float*)d_in[7];
  const float* D_par  = (const float*)d_in[8];
  const float* W_out  = (const float*)d_in[9];
  float* out = (float*)d_out;

  // Workspace (floats): proj | uc | xdbc | dt | yg | Ap | Bp   (~113 MB)
  float* ws   = (float*)d_ws;
  float* proj = ws;                                       // SEQ * 2*D_INNER
  float* uc   = proj + (size_t)SEQ * 2 * D_INNER;         // SEQ * D_INNER
  float* xdbc = uc   + (size_t)SEQ * D_INNER;             // SEQ * XDBC_COLS
  float* dtb  = xdbc + (size_t)SEQ * XDBC_COLS;           // SEQ * D_INNER
  float* yg   = dtb  + (size_t)SEQ * D_INNER;             // SEQ * D_INNER
  float* Ap   = yg   + (size_t)SEQ * D_INNER;             // <= SEQ * D_INNER
  float* Bp   = Ap   + (size_t)SEQ * D_INNER;             // <= D_MODEL * 2*D_INNER

  const dim3 wave(32);
  const int mb = SEQ / 32;   // 64 row-blocks of 32

  auto packA = [&](const float* A, int lda, int M, int K) {
    pack_a_kernel<<<((size_t)M * K) / 256, 256, 0, stream>>>(A, lda, Ap, K);
  };
  auto packB = [&](const float* B, int N, int K) {
    pack_b_kernel<<<((size_t)N * K) / 256, 256, 0, stream>>>(B, N, Bp, K);
  };

  // 1) proj = x @ W_in                        (2048x1024)@(1024x4096)
  packA(x, D_MODEL, SEQ, D_MODEL);
  packB(W_in, 2 * D_INNER, D_MODEL);
  gemm_wmma_packed<4, 0><<<dim3((2 * D_INNER) / 64, mb), wave, 0, stream>>>(
      Ap, Bp, proj, 2 * D_INNER, D_MODEL, 0, nullptr);

  // 2) uc = silu(causal_dwconv(u))
  conv_silu_kernel<<<(SEQ * D_INNER) / 256, 256, 0, stream>>>(
      proj, conv_w, conv_b, uc);

  // 3) xdbc = uc @ W_x                        (2048x2048)@(2048x160)
  packA(uc, D_INNER, SEQ, D_INNER);
  packB(W_x, XDBC_COLS, D_INNER);
  gemm_wmma_packed<4, 0><<<dim3(2, mb), wave, 0, stream>>>(
      Ap, Bp, xdbc, XDBC_COLS, D_INNER, 0, nullptr);
  gemm_wmma_packed<2, 0><<<dim3(1, mb), wave, 0, stream>>>(
      Ap, Bp, xdbc, XDBC_COLS, D_INNER, 8, nullptr);

  // 4) dt = softplus(dt_r @ W_dt + b_dt)      dt_r = xdbc[:, :128] (lda=160)
  packA(xdbc, XDBC_COLS, SEQ, DT_RANK);
  packB(W_dt, D_INNER, DT_RANK);
  gemm_wmma_packed<4, 1><<<dim3(D_INNER / 64, mb), wave, 0, stream>>>(
      Ap, Bp, dtb, D_INNER, DT_RANK, 0, b_dt);

  // 5) selective scan + D skip + silu(z) gate
  scan_kernel<<<D_INNER / 16, 256, 0, stream>>>(
      uc, dtb, xdbc, A_log, D_par, proj, yg);

  // 6) out = yg @ W_out                       (2048x2048)@(2048x1024)
  packA(yg, D_INNER, SEQ, D_INNER);
  packB(W_out, D_MODEL, D_INNER);
  gemm_wmma_packed<4, 0><<<dim3(D_MODEL / 64, mb), wave, 0, stream>>>(
      Ap, Bp, out, D_MODEL, D_INNER, 0, nullptr);
}